// FraudDetector_42949672960530
// MI455X (gfx1250) — compile-verified
//
#include <hip/hip_runtime.h>
#include <stdint.h>

typedef __attribute__((ext_vector_type(2))) float v2f;
typedef __attribute__((ext_vector_type(8))) float v8f;

#define HIDDEN 512
#define LDS_STRIDE 516  // 512 + 4 floats pad -> conflict-free A-fragment reads

// ---------------- degree / normalization ----------------

__global__ void deg_init_kernel(float* __restrict__ deg, int n) {
    int i = blockIdx.x * blockDim.x + threadIdx.x;
    if (i < n) deg[i] = 1.0f;  // self-loop
}

__global__ void deg_accum_kernel(const int* __restrict__ dst, float* __restrict__ deg, int e) {
    int i = blockIdx.x * blockDim.x + threadIdx.x;
    if (i < e) atomicAdd(&deg[dst[i]], 1.0f);
}

__global__ void deg_rsqrt_kernel(float* __restrict__ deg, int n) {
    int i = blockIdx.x * blockDim.x + threadIdx.x;
    if (i < n) deg[i] = rsqrtf(deg[i]);  // deg >= 1 thanks to self-loops
}

// ---------------- WMMA GEMM: H = X (n x 512) * W (512 x 512) ----------------
// Block: 256 threads = 8 waves. Block owns 16 rows, all 512 cols.
// Each wave: 16x64 output tile = 4 accumulators of V_WMMA_F32_16X16X4_F32.
__global__ __launch_bounds__(256) void gemm_wmma_kernel(const float* __restrict__ X,
                                                        const float* __restrict__ W,
                                                        float* __restrict__ H, int n) {
    __shared__ float xs[16 * LDS_STRIDE];
    const int tid  = threadIdx.x;
    const int row0 = blockIdx.x * 16;

    // Cooperative stage of the 16x512 X strip into padded LDS (float4 per load)
    #pragma unroll
    for (int t = 0; t < 8; ++t) {
        int idx = tid + t * 256;         // 0..2047
        int r   = idx >> 7;              // 0..15
        int c4  = (idx & 127) << 2;      // 0..508 step 4
        if (row0 + r < n) {
            float4 v = *(const float4*)(X + (size_t)(row0 + r) * HIDDEN + c4);
            *(float4*)(xs + r * LDS_STRIDE + c4) = v;
        }
    }
    __syncthreads();

    const int lane    = tid & 31;
    const int wid     = tid >> 5;        // 0..7
    const int colbase = wid * 64;
    const int hi      = lane >> 4;       // 0 (lanes 0-15) / 1 (lanes 16-31)
    const int m       = lane & 15;       // M index (A) / N index (B)
    const int kk      = hi * 2;          // lane-half K pair select

    v8f acc0 = {}, acc1 = {}, acc2 = {}, acc3 = {};
    const float* xrow = xs + m * LDS_STRIDE + kk;

    #pragma unroll 4
    for (int k = 0; k < HIDDEN; k += 4) {
        // A fragment (16x4 f32): V0 = x[m][k+kk], V1 = x[m][k+kk+1]
        v2f a;
        a.x = xrow[k];
        a.y = xrow[k + 1];
        // B fragments (4x16 f32): V0 = W[k+kk][col], V1 = W[k+kk+1][col]
        const float* wp = W + (size_t)(k + kk) * HIDDEN + colbase + m;
        v2f b0, b1, b2, b3;
        b0.x = wp[0];   b0.y = wp[HIDDEN + 0];
        b1.x = wp[16];  b1.y = wp[HIDDEN + 16];
        b2.x = wp[32];  b2.y = wp[HIDDEN + 32];
        b3.x = wp[48];  b3.y = wp[HIDDEN + 48];
        acc0 = __builtin_amdgcn_wmma_f32_16x16x4_f32(false, a, false, b0, (short)0, acc0, false, false);
        acc1 = __builtin_amdgcn_wmma_f32_16x16x4_f32(false, a, false, b1, (short)0, acc1, false, false);
        acc2 = __builtin_amdgcn_wmma_f32_16x16x4_f32(false, a, false, b2, (short)0, acc2, false, false);
        acc3 = __builtin_amdgcn_wmma_f32_16x16x4_f32(false, a, false, b3, (short)0, acc3, false, false);
    }

    // C/D layout: VGPR r, lanes 0-15 -> M=r, lanes 16-31 -> M=r+8; N = lane&15
    const int srow = row0 + (hi << 3);
    float* hp = H + (size_t)srow * HIDDEN + colbase + m;
    #pragma unroll
    for (int r = 0; r < 8; ++r) {
        if (srow + r < n) {
            hp[(size_t)r * HIDDEN + 0]  = acc0[r];
            hp[(size_t)r * HIDDEN + 16] = acc1[r];
            hp[(size_t)r * HIDDEN + 32] = acc2[r];
            hp[(size_t)r * HIDDEN + 48] = acc3[r];
        }
    }
}

// ---------------- agg = bias + h * dis^2 (self-loop), full overwrite ----------------
__global__ void init_agg_kernel(const float* __restrict__ H, const float* __restrict__ bias,
                                const float* __restrict__ dis, float* __restrict__ agg,
                                long long total) {
    long long gid = (long long)blockIdx.x * blockDim.x + threadIdx.x;
    if (gid >= total) return;
    int row = (int)(gid >> 7);
    int c4  = ((int)gid & 127) << 2;
    float w = dis[row]; w *= w;
    float4 hv = *(const float4*)(H + (size_t)row * HIDDEN + c4);
    float4 bv = *(const float4*)(bias + c4);
    float4 o;
    o.x = bv.x + hv.x * w;
    o.y = bv.y + hv.y * w;
    o.z = bv.z + hv.z * w;
    o.w = bv.w + hv.w * w;
    *(float4*)(agg + (size_t)row * HIDDEN + c4) = o;
}

// ---------------- edge scatter: agg[dst] += h[src] * dis[src]*dis[dst] ----------------
__global__ void scatter_kernel(const float* __restrict__ H, const int* __restrict__ src,
                               const int* __restrict__ dst, const float* __restrict__ dis,
                               float* __restrict__ agg, long long total) {
    long long gid = (long long)blockIdx.x * blockDim.x + threadIdx.x;
    if (gid >= total) return;
    int e  = (int)(gid >> 7);
    int c4 = ((int)gid & 127) << 2;
    int s = src[e];
    int d = dst[e];
    float w = dis[s] * dis[d];
    float4 hv = *(const float4*)(H + (size_t)s * HIDDEN + c4);
    float* ap = agg + (size_t)d * HIDDEN + c4;
    atomicAdd(ap + 0, hv.x * w);
    atomicAdd(ap + 1, hv.y * w);
    atomicAdd(ap + 2, hv.z * w);
    atomicAdd(ap + 3, hv.w * w);
}

// ---------------- in-place relu ----------------
__global__ void relu_kernel(float* __restrict__ a, long long count4) {
    long long gid = (long long)blockIdx.x * blockDim.x + threadIdx.x;
    if (gid >= count4) return;
    float4 v = *(float4*)(a + gid * 4);
    v.x = fmaxf(v.x, 0.0f);
    v.y = fmaxf(v.y, 0.0f);
    v.z = fmaxf(v.z, 0.0f);
    v.w = fmaxf(v.w, 0.0f);
    *(float4*)(a + gid * 4) = v;
}

extern "C" void kernel_launch(void* const* d_in, const int* in_sizes, int n_in,
                              void* d_out, int out_size, void* d_ws, size_t ws_size,
                              hipStream_t stream) {
    const float* x    = (const float*)d_in[0];
    const int*   eidx = (const int*)d_in[1];   // [2, E] int
    const float* W    = (const float*)d_in[2]; // [L, 512, 512]
    const float* b    = (const float*)d_in[3]; // [L, 512]

    const int n      = in_sizes[0] / HIDDEN;
    const int e      = in_sizes[1] / 2;
    const int layers = in_sizes[2] / (HIDDEN * HIDDEN);
    const int* srcp = eidx;
    const int* dstp = eidx + e;

    // workspace layout: dis | H | A  (~205 MB)
    char*  ws   = (char*)d_ws;
    float* dis  = (float*)ws;
    size_t off  = ((size_t)n * sizeof(float) + 255) & ~(size_t)255;
    float* H    = (float*)(ws + off);
    size_t feat = (size_t)n * HIDDEN * sizeof(float);
    float* A    = (float*)(ws + off + feat);

    // symmetric normalization: dis[i] = rsqrt(1 + indegree)
    deg_init_kernel <<<(n + 255) / 256, 256, 0, stream>>>(dis, n);
    deg_accum_kernel<<<(e + 255) / 256, 256, 0, stream>>>(dstp, dis, e);
    deg_rsqrt_kernel<<<(n + 255) / 256, 256, 0, stream>>>(dis, n);

    const long long nt_feat = (long long)n * 128;       // (n*512)/4 threads
    const long long nt_edge = (long long)e * 128;       // (e*512)/4 threads
    const long long c4      = (long long)n * HIDDEN / 4;
    const int g_feat = (int)((nt_feat + 255) / 256);
    const int g_edge = (int)((nt_edge + 255) / 256);
    const int g_relu = (int)((c4 + 255) / 256);
    const int g_gemm = (n + 15) / 16;

    const float* xin = x;
    for (int l = 0; l < layers; ++l) {
        float* agg = (l == layers - 1) ? (float*)d_out : A;
        gemm_wmma_kernel<<<g_gemm, 256, 0, stream>>>(xin, W + (size_t)l * HIDDEN * HIDDEN, H, n);
        init_agg_kernel <<<g_feat, 256, 0, stream>>>(H, b + (size_t)l * HIDDEN, dis, agg, nt_feat);
        scatter_kernel  <<<g_edge, 256, 0, stream>>>(H, srcp, dstp, dis, agg, nt_edge);
        relu_kernel     <<<g_relu, 256, 0, stream>>>(agg, c4);
        xin = agg;  // layer l output becomes next input (A reused: read fully by GEMM first)
    }
}